// _AnchorTargetLayer_66752381714838
// MI455X (gfx1250) — compile-verified
//
#include <hip/hip_runtime.h>
#include <hip/hip_bf16.h>

typedef __attribute__((ext_vector_type(2))) float v2f;
typedef __attribute__((ext_vector_type(8))) float v8f;

#define A_TOTAL   49152      // 12 poses * 64*64 cells
#define B_TOTAL   16
#define FEAT      64
#define RATIO     4.0f       // 256 / 64
#define NKPT      21
#define NBH       32         // B * 2 hands

// ---- workspace layout (floats) ----
#define WS_BASE   0          // 12*21*2 = 504   base anchor poses
#define WS_GX     504        // 32*21 = 672     gt x
#define WS_GY     1176       // 672             gt y
#define WS_GN     1848       // 672             gx^2+gy^2
#define WS_VIS    2520       // 672             visibility (0/1)
#define WS_INVS   3192       // 32              1/(2*s2*k2)
#define WS_CNT    3224       // 32              max(sum vis,1)
#define WS_HT     3256       // 32              hand_type mask per (b,h)

// ---- output layout (floats) ----
#define OUT_ANCH  0                       // A*21*2 = 2064384
#define OUT_INS   2064384                 // A      = 49152
#define OUT_LAB   (2064384 + 49152)       // B*A*3  = 2359296
#define OUT_OFF   (2064384 + 49152 + 2359296) // B*A*21*2 = 33030144

__constant__ float c_ref[NKPT][2] = {
    {213.33335876464844f, 124.50563049316406f}, {190.504638671875f, 115.11840057373047f},
    {169.9791717529297f, 101.77180480957031f},  {146.72341918945312f, 96.25749206542969f},
    {128.86770629882812f, 87.2344970703125f},   {150.34292602539062f, 101.61070251464844f},
    {119.29926300048828f, 98.73982238769531f},  {100.03463745117188f, 99.74459838867188f},
    {82.62400817871094f, 101.2509536743164f},   {148.91049194335938f, 112.71517181396484f},
    {114.37303161621094f, 113.20121002197266f}, {91.90096282958984f, 116.49812316894531f},
    {74.75020599365234f, 119.37875366210938f},  {149.59658813476562f, 124.09295654296875f},
    {119.72419738769531f, 126.36898040771484f}, {99.59107208251953f, 129.40196228027344f},
    {82.82524108886719f, 131.584228515625f},    {154.55911254882812f, 135.07681274414062f},
    {133.8833770751953f, 140.85983276367188f},  {120.45906066894531f, 145.40306091308594f},
    {106.21541595458984f, 150.072265625f}};
__constant__ float c_scales[3] = {0.25f, 0.5f, 1.0f};

// ------------------------------------------------------------------
// Kernel 1: base poses + gt-derived tables into workspace
// ------------------------------------------------------------------
__global__ __launch_bounds__(256) void setup_kernel(const float* __restrict__ pose_gt,
                                                    const int* __restrict__ hand_type,
                                                    float* __restrict__ ws) {
    int tid = threadIdx.x;
    // center of reference pose
    float cx = 0.f, cy = 0.f;
    for (int k = 0; k < NKPT; ++k) { cx += c_ref[k][0]; cy += c_ref[k][1]; }
    cx *= (1.0f / NKPT); cy *= (1.0f / NKPT);

    // base anchor poses: 12 = 3 scales x 4 orientations (exact rotations)
    for (int t = tid; t < 12 * NKPT; t += 256) {
        int p = t / NKPT, k = t % NKPT;
        float s = c_scales[p >> 2];
        int o = p & 3;
        float vx = c_ref[k][0] - cx, vy = c_ref[k][1] - cy;
        float rx, ry;
        if (o == 0)      { rx =  vx; ry =  vy; }
        else if (o == 1) { rx = -vy; ry =  vx; }
        else if (o == 2) { rx = -vx; ry = -vy; }
        else             { rx =  vy; ry = -vx; }
        ws[WS_BASE + t * 2 + 0] = cx + s * rx;
        ws[WS_BASE + t * 2 + 1] = cy + s * ry;
    }
    // gt tables
    for (int t = tid; t < NBH * NKPT; t += 256) {
        int bh = t / NKPT, k = t % NKPT;
        int b = bh >> 1, h = bh & 1;
        const float* g = pose_gt + ((b * 42) + h * NKPT + k) * 3;
        float gx = g[0], gy = g[1];
        float v = (g[2] > 0.f) ? 1.f : 0.f;
        ws[WS_GX  + t] = gx;
        ws[WS_GY  + t] = gy;
        ws[WS_GN  + t] = gx * gx + gy * gy;
        ws[WS_VIS + t] = v;
    }
    __syncthreads();
    // per-(b,h) scalars
    if (tid < NBH) {
        int bh = tid, b = bh >> 1, h = bh & 1;
        float mnx = 3.4e38f, mxx = -3.4e38f, mny = 3.4e38f, mxy = -3.4e38f, cv = 0.f;
        for (int k = 0; k < NKPT; ++k) {
            float x = ws[WS_GX + bh * NKPT + k], y = ws[WS_GY + bh * NKPT + k];
            mnx = fminf(mnx, x); mxx = fmaxf(mxx, x);
            mny = fminf(mny, y); mxy = fmaxf(mxy, y);
            cv += ws[WS_VIS + bh * NKPT + k];
        }
        float area = (mxx - mnx) * (mxy - mny);
        float s2 = fmaxf(area, 1.0f);
        ws[WS_INVS + bh] = 1.0f / (2.0f * s2 * 0.01f);   // k2 = (2*0.05)^2 = 0.01
        ws[WS_CNT  + bh] = fmaxf(cv, 1.0f);
        ws[WS_HT   + bh] = (hand_type[b * 2 + h] == 1) ? 1.f : 0.f;
    }
}

// ------------------------------------------------------------------
// Kernel 2: anchors + inside flags (streaming)
// ------------------------------------------------------------------
__global__ __launch_bounds__(256) void anchors_kernel(const float* __restrict__ ws,
                                                      float* __restrict__ anchors,
                                                      float* __restrict__ insideF) {
    int a = blockIdx.x * 256 + threadIdx.x;
    if (a >= A_TOTAL) return;
    int cell = a / 12, pose = a % 12;
    float shx = (float)(cell & (FEAT - 1)) * RATIO;
    float shy = (float)(cell >> 6) * RATIO;
    bool ins = true;
    #pragma unroll 3
    for (int k = 0; k < NKPT; ++k) {
        float x = ws[WS_BASE + (pose * NKPT + k) * 2 + 0] + shx;
        float y = ws[WS_BASE + (pose * NKPT + k) * 2 + 1] + shy;
        anchors[(a * NKPT + k) * 2 + 0] = x;
        anchors[(a * NKPT + k) * 2 + 1] = y;
        ins = ins && (x >= 0.f) && (y >= 0.f) && (x < 256.f) && (y < 256.f);
    }
    insideF[a] = ins ? 1.f : 0.f;
}

// ------------------------------------------------------------------
// Kernel 3: OKS similarity via f32 WMMA + label assignment.
// Per wave: 16 anchors x 32 (b,h) columns.
// d^2 = A(16x4) * B(4x16): A row = (ax, ay, |a|^2, 1), B col = (-2gx, -2gy, 1, |g|^2)
// ------------------------------------------------------------------
__global__ __launch_bounds__(256) void sim_labels_kernel(const float* __restrict__ ws,
                                                         const float* __restrict__ insideF,
                                                         float* __restrict__ labels) {
    __shared__ float s_base[12 * NKPT * 2];
    __shared__ float s_gx[NBH * NKPT], s_gy[NBH * NKPT], s_gn[NBH * NKPT], s_vis[NBH * NKPT];
    __shared__ float s_invs[NBH], s_cnt[NBH], s_ht[NBH];
    int tid = threadIdx.x;
    for (int i = tid; i < 12 * NKPT * 2; i += 256) s_base[i] = ws[WS_BASE + i];
    for (int i = tid; i < NBH * NKPT; i += 256) {
        s_gx[i] = ws[WS_GX + i]; s_gy[i] = ws[WS_GY + i];
        s_gn[i] = ws[WS_GN + i]; s_vis[i] = ws[WS_VIS + i];
    }
    if (tid < NBH) {
        s_invs[tid] = ws[WS_INVS + tid];
        s_cnt[tid]  = ws[WS_CNT + tid];
        s_ht[tid]   = ws[WS_HT + tid];
    }
    __syncthreads();

    int wave = tid >> 5, lane = tid & 31;
    int tile = blockIdx.x * 8 + wave;          // one 16-anchor tile per wave
    int aBase = tile * 16;
    int lrow  = lane & 15;
    bool hiK  = lane >= 16;                    // A: lanes 16-31 carry K=2,3

    // A-matrix anchor for this lane: row M = lrow
    int aA = aBase + lrow;
    int cell = aA / 12, pose = aA % 12;
    float shx = (float)(cell & (FEAT - 1)) * RATIO;
    float shy = (float)(cell >> 6) * RATIO;

    int col0 = lrow;          // (b,h) column for tile 0
    int col1 = 16 + lrow;     // (b,h) column for tile 1
    float invs0 = s_invs[col0], invs1 = s_invs[col1];

    v8f sim0 = {}; v8f sim1 = {};
    v8f zero = {};
    for (int k = 0; k < NKPT; ++k) {
        float ax = s_base[(pose * NKPT + k) * 2 + 0] + shx;
        float ay = s_base[(pose * NKPT + k) * 2 + 1] + shy;
        v2f Aop;
        Aop.x = hiK ? (ax * ax + ay * ay) : ax;   // K2 = |a|^2 ; K0 = ax
        Aop.y = hiK ? 1.0f : ay;                  // K3 = 1     ; K1 = ay
        float gx0 = s_gx[col0 * NKPT + k], gy0 = s_gy[col0 * NKPT + k], gn0 = s_gn[col0 * NKPT + k];
        float gx1 = s_gx[col1 * NKPT + k], gy1 = s_gy[col1 * NKPT + k], gn1 = s_gn[col1 * NKPT + k];
        v2f B0, B1;
        B0.x = hiK ? 1.0f : (-2.0f * gx0);        // K2 = 1     ; K0 = -2gx
        B0.y = hiK ? gn0  : (-2.0f * gy0);        // K3 = |g|^2 ; K1 = -2gy
        B1.x = hiK ? 1.0f : (-2.0f * gx1);
        B1.y = hiK ? gn1  : (-2.0f * gy1);
        v8f d0 = __builtin_amdgcn_wmma_f32_16x16x4_f32(false, Aop, false, B0, (short)0, zero, false, false);
        v8f d1 = __builtin_amdgcn_wmma_f32_16x16x4_f32(false, Aop, false, B1, (short)0, zero, false, false);
        float v0 = s_vis[col0 * NKPT + k], v1 = s_vis[col1 * NKPT + k];
        #pragma unroll
        for (int j = 0; j < 8; ++j) {
            sim0[j] += v0 * __expf(-d0[j] * invs0);
            sim1[j] += v1 * __expf(-d1[j] * invs1);
        }
    }
    float c0 = s_cnt[col0], c1 = s_cnt[col1];
    bool evenl = (lane & 1) == 0;
    int b0 = col0 >> 1, b1 = col1 >> 1;
    float ht0r = s_ht[b0 * 2 + 0], ht0l = s_ht[b0 * 2 + 1];
    float ht1r = s_ht[b1 * 2 + 0], ht1l = s_ht[b1 * 2 + 1];

    #pragma unroll
    for (int j = 0; j < 8; ++j) {
        int M = hiK ? (j + 8) : j;             // C/D row layout
        int anc = aBase + M;
        bool ins = insideF[anc] > 0.5f;
        // ---- group 0 (b = 0..7) ----
        {
            float m = sim0[j] / c0;
            float o = __shfl_xor(m, 1, 32);    // partner hand column
            float s_r = evenl ? m : o;
            float s_l = evenl ? o : m;
            float mx = fmaxf(s_r, s_l);
            bool am0 = (s_r >= s_l);           // argmax ties -> index 0 (right)
            bool rF = am0 && (mx > 0.5f) && (ht0r > 0.5f) && ins;
            bool lF = !am0 && (mx > 0.5f) && (ht0l > 0.5f) && ins;
            bool nF = !rF && !lF && ins;
            if (evenl) {
                float* L = labels + ((size_t)b0 * A_TOTAL + anc) * 3;
                L[0] = ins ? (lF ? 1.f : 0.f) : -1.f;
                L[1] = ins ? (nF ? 1.f : 0.f) : -1.f;
                L[2] = ins ? (rF ? 1.f : 0.f) : -1.f;
            }
        }
        // ---- group 1 (b = 8..15) ----
        {
            float m = sim1[j] / c1;
            float o = __shfl_xor(m, 1, 32);
            float s_r = evenl ? m : o;
            float s_l = evenl ? o : m;
            float mx = fmaxf(s_r, s_l);
            bool am0 = (s_r >= s_l);
            bool rF = am0 && (mx > 0.5f) && (ht1r > 0.5f) && ins;
            bool lF = !am0 && (mx > 0.5f) && (ht1l > 0.5f) && ins;
            bool nF = !rF && !lF && ins;
            if (evenl) {
                float* L = labels + ((size_t)b1 * A_TOTAL + anc) * 3;
                L[0] = ins ? (lF ? 1.f : 0.f) : -1.f;
                L[1] = ins ? (nF ? 1.f : 0.f) : -1.f;
                L[2] = ins ? (rF ? 1.f : 0.f) : -1.f;
            }
        }
    }
}

// ------------------------------------------------------------------
// Kernel 4: offsets (bandwidth-dominant pass, coalesced float2 stores)
// ------------------------------------------------------------------
__global__ __launch_bounds__(256) void offsets_kernel(const float* __restrict__ ws,
                                                      const float* __restrict__ anchors,
                                                      const float* __restrict__ labels,
                                                      float* __restrict__ offs) {
    int idx = blockIdx.x * 256 + threadIdx.x;          // (b*A + a)*21 + k
    if (idx >= B_TOTAL * A_TOTAL * NKPT) return;
    int k = idx % NKPT;
    int ba = idx / NKPT;
    int a = ba % A_TOTAL;
    int b = ba / A_TOTAL;
    const float* L = labels + (size_t)ba * 3;
    float l0 = L[0];
    bool inside = (l0 >= 0.f);
    bool leftF  = (l0 > 0.5f);
    bool rightF = (L[2] > 0.5f);
    float ox = 0.f, oy = 0.f;
    if (inside) {
        float tx = 0.f, ty = 0.f;
        if (rightF) {
            tx = ws[WS_GX + (b * 2 + 0) * NKPT + k];
            ty = ws[WS_GY + (b * 2 + 0) * NKPT + k];
        } else if (leftF) {
            tx = ws[WS_GX + (b * 2 + 1) * NKPT + k];
            ty = ws[WS_GY + (b * 2 + 1) * NKPT + k];
        }
        ox = tx - anchors[(a * NKPT + k) * 2 + 0];
        oy = ty - anchors[(a * NKPT + k) * 2 + 1];
    }
    offs[(size_t)idx * 2 + 0] = ox;
    offs[(size_t)idx * 2 + 1] = oy;
}

extern "C" void kernel_launch(void* const* d_in, const int* in_sizes, int n_in,
                              void* d_out, int out_size, void* d_ws, size_t ws_size,
                              hipStream_t stream) {
    const float* pose_gt   = (const float*)d_in[0];
    const int*   hand_type = (const int*)d_in[1];
    // d_in[2]/d_in[3] are feat_h/feat_w scalars; grid geometry is host-side -> 64 (per setup).
    float* out = (float*)d_out;
    float* anchors = out + OUT_ANCH;
    float* insideF = out + OUT_INS;
    float* labels  = out + OUT_LAB;
    float* offs    = out + OUT_OFF;
    float* ws = (float*)d_ws;

    setup_kernel<<<1, 256, 0, stream>>>(pose_gt, hand_type, ws);
    anchors_kernel<<<A_TOTAL / 256, 256, 0, stream>>>(ws, anchors, insideF);
    sim_labels_kernel<<<A_TOTAL / (16 * 8), 256, 0, stream>>>(ws, insideF, labels);
    offsets_kernel<<<(B_TOTAL * A_TOTAL * NKPT) / 256, 256, 0, stream>>>(ws, anchors, labels, offs);
}